// SpanRepresentation_32487132627590
// MI455X (gfx1250) — compile-verified
//
#include <hip/hip_runtime.h>
#include <cstdint>
#include <cstddef>

#define S_LEN  2048
#define D_DIM  768
#define NCHUNK 32
#define CHUNK  64    // S_LEN / NCHUNK
#define PW     20
#define OUTW   2364  // 3*768 + 3*20
#define NSLOT  591   // OUTW / 4 (every segment boundary is 16B-aligned)

// ---------------------------------------------------------------------------
// Kernel 1: per-column partial sums over chunks of the sequence axis.
// 192 threads, each owns a float4 column strip -> all traffic is b128.
// grid = (NCHUNK, B)
// ---------------------------------------------------------------------------
__global__ __launch_bounds__(192) void ps_kernel(const float* __restrict__ x,
                                                 float* __restrict__ part) {
    const int t = threadIdx.x;                 // 0..191 -> columns 4t..4t+3
    const int c = blockIdx.x;
    const int b = blockIdx.y;
    const float4* p = reinterpret_cast<const float4*>(
        x + ((size_t)(b * S_LEN + c * CHUNK)) * D_DIM) + t;
    float4 acc = make_float4(0.f, 0.f, 0.f, 0.f);
#pragma unroll 8
    for (int i = 0; i < CHUNK; ++i) {
        const float4 v = p[(size_t)i * (D_DIM / 4)];
        acc.x += v.x; acc.y += v.y; acc.z += v.z; acc.w += v.w;
    }
    reinterpret_cast<float4*>(part + ((size_t)(b * NCHUNK + c)) * D_DIM)[t] = acc;
}

// ---------------------------------------------------------------------------
// Kernel 2: scan chunk partials, then serially accumulate inside the chunk,
// writing the inclusive cumsum cs[b, s, d]. b128 loads/stores + prefetch.
// grid = (NCHUNK, B)
// ---------------------------------------------------------------------------
__global__ __launch_bounds__(192) void scan_kernel(const float* __restrict__ x,
                                                   const float* __restrict__ part,
                                                   float* __restrict__ cs) {
    const int t = threadIdx.x;
    const int c = blockIdx.x;
    const int b = blockIdx.y;
    float4 acc = make_float4(0.f, 0.f, 0.f, 0.f);
    const float4* pp = reinterpret_cast<const float4*>(
        part + ((size_t)b * NCHUNK) * D_DIM) + t;
    for (int cc = 0; cc < c; ++cc) {
        const float4 v = pp[(size_t)cc * (D_DIM / 4)];
        acc.x += v.x; acc.y += v.y; acc.z += v.z; acc.w += v.w;
    }
    const float4* p = reinterpret_cast<const float4*>(
        x + ((size_t)(b * S_LEN + c * CHUNK)) * D_DIM) + t;
    float4* q = reinterpret_cast<float4*>(
        cs + ((size_t)(b * S_LEN + c * CHUNK)) * D_DIM) + t;
    for (int i = 0; i < CHUNK; ++i) {
        __builtin_prefetch(p + (size_t)(i + 8) * (D_DIM / 4), 0, 0); // global_prefetch_b8
        const float4 v = p[(size_t)i * (D_DIM / 4)];
        acc.x += v.x; acc.y += v.y; acc.z += v.z; acc.w += v.w;
        q[(size_t)i * (D_DIM / 4)] = acc;
    }
}

// ---------------------------------------------------------------------------
// CDNA5 async global -> LDS copy, 16 bytes per lane, 64-bit vaddr form.
// LDS byte address comes from the lds_off VGPR; tracked by ASYNCcnt.
// ---------------------------------------------------------------------------
__device__ __forceinline__ void async_b128(unsigned lds_off, unsigned long long gaddr) {
    asm volatile("global_load_async_to_lds_b128 %0, %1, off"
                 :: "v"(lds_off), "v"(gaddr)
                 : "memory");
}

__device__ __forceinline__ void wait_async0() {
    asm volatile("s_wait_asynccnt 0" ::: "memory");
}

// ---------------------------------------------------------------------------
// Kernel 3: one block per (b, n) span. Stage the 4 gathered rows into LDS via
// async loads; emit the 2364-float row as 591 float4 stores.
// LDS rows: [cs[s1] | cs[s0-1] | x[s0] | x[s1]], each D_DIM floats.
// Slot map (f = output_float4_index, store at orow + 4f):
//   [0,192)   mean   <- (sm0 - sel*sm1)/width
//   [192,384) x[s0]  <- sm2
//   [384,389) pos0   <- ptab[l0]
//   [389,581) x[s1]  <- sm3
//   [581,586) pos1   <- ptab[l1]
//   [586,591) width  <- wtab[em]
// ---------------------------------------------------------------------------
__global__ __launch_bounds__(256) void gather_kernel(
    const float* __restrict__ x,     const int* __restrict__ spans,
    const int* __restrict__ ptl,     const float* __restrict__ wtab,
    const float* __restrict__ ptab,  const float* __restrict__ cs,
    float* __restrict__ out,         int N) {
    __shared__ __align__(16) float sm[4 * D_DIM];

    const int n   = blockIdx.x;
    const int b   = blockIdx.y;
    const int tid = threadIdx.x;

    const int s0 = spans[2 * n];
    const int s1 = spans[2 * n + 1];
    const int width = s1 - s0 + 1;
    const int s0m = (s0 > 0) ? (s0 - 1) : 0;

    const float* pcs1 = cs + ((size_t)(b * S_LEN + s1))  * D_DIM;
    const float* pcs0 = cs + ((size_t)(b * S_LEN + s0m)) * D_DIM;
    const float* px0  = x  + ((size_t)(b * S_LEN + s0))  * D_DIM;
    const float* px1  = x  + ((size_t)(b * S_LEN + s1))  * D_DIM;

    // Low 32 bits of a flat LDS pointer are the LDS byte offset.
    const unsigned ldsbase = (unsigned)(uintptr_t)&sm[0];

    // 4 rows * 3072 B = 768 lane-slots of 16 B; 3 passes of 256 lanes.
    // 192 slots/row and 192 % 32 == 0 -> each wave's row is uniform, coalesced.
#pragma unroll
    for (int p = 0; p < 3; ++p) {
        const int f = p * 256 + tid;       // 0..767
        const int r = f / 192;             // source row
        const int e = f - r * 192;         // 16B element within row
        const float* bp = (r == 0) ? pcs1 : (r == 1) ? pcs0 : (r == 2) ? px0 : px1;
        async_b128(ldsbase + (unsigned)(r * 3072 + e * 16),
                   (unsigned long long)(uintptr_t)bp + (unsigned long long)(e * 16));
    }
    wait_async0();      // this wave's async LDS writes have landed
    __syncthreads();    // all waves'

    // em = searchsorted(BINS, width, 'right') - 1  (branchless)
    const int bins[16] = {0,1,2,3,4,5,7,8,9,10,15,16,31,32,63,64};
    int em = -1;
#pragma unroll
    for (int i = 0; i < 16; ++i) em += (width >= bins[i]) ? 1 : 0;

    const int   l0   = ptl[s0];
    const int   l1   = ptl[s1];
    const float wf   = (float)width;
    const float csel = (s0 > 0) ? 1.0f : 0.0f;   // cs[-1] contributes 0

    float* orow = out + ((size_t)b * N + n) * OUTW;   // row base 16B-aligned

#pragma unroll
    for (int p = 0; p < 3; ++p) {
        const int f = p * 256 + tid;
        if (f < NSLOT) {
            float4 v;
            if (f < 192) {                                   // mean
                const float4 c1 = *reinterpret_cast<const float4*>(sm + 4 * f);
                const float4 c0 = *reinterpret_cast<const float4*>(sm + D_DIM + 4 * f);
                v.x = (c1.x - csel * c0.x) / wf;
                v.y = (c1.y - csel * c0.y) / wf;
                v.z = (c1.z - csel * c0.z) / wf;
                v.w = (c1.w - csel * c0.w) / wf;
            } else if (f < 384) {                            // x[s0]
                v = *reinterpret_cast<const float4*>(sm + 2 * D_DIM + 4 * (f - 192));
            } else if (f < 389) {                            // pos_emb0
                v = *reinterpret_cast<const float4*>(ptab + l0 * PW + 4 * (f - 384));
            } else if (f < 581) {                            // x[s1]
                v = *reinterpret_cast<const float4*>(sm + 3 * D_DIM + 4 * (f - 389));
            } else if (f < 586) {                            // pos_emb1
                v = *reinterpret_cast<const float4*>(ptab + l1 * PW + 4 * (f - 581));
            } else {                                         // width_emb
                v = *reinterpret_cast<const float4*>(wtab + em * PW + 4 * (f - 586));
            }
            *reinterpret_cast<float4*>(orow + 4 * f) = v;    // global_store_b128
        }
    }
}

// ---------------------------------------------------------------------------
extern "C" void kernel_launch(void* const* d_in, const int* in_sizes, int n_in,
                              void* d_out, int out_size, void* d_ws, size_t ws_size,
                              hipStream_t stream) {
    (void)n_in; (void)out_size; (void)ws_size;
    const float* x     = (const float*)d_in[0];   // (B, S, D) f32
    const int*   spans = (const int*)  d_in[1];   // (N, 2)    i32
    const int*   ptl   = (const int*)  d_in[2];   // (S,)      i32
    const float* wtab  = (const float*)d_in[3];   // (16, 20)  f32
    const float* ptab  = (const float*)d_in[4];   // (29, 20)  f32
    float*       out   = (float*)d_out;

    const int B = in_sizes[0] / (S_LEN * D_DIM);  // 4
    const int N = in_sizes[1] / 2;                // 4096

    // Workspace: inclusive cumsum cs (B*S*D f32 = 25.2 MB), then partials.
    float* cs   = (float*)d_ws;
    float* part = cs + (size_t)B * S_LEN * D_DIM;

    dim3 scanGrid(NCHUNK, B);                     // (32, B), 192 threads
    ps_kernel  <<<scanGrid, 192, 0, stream>>>(x, part);
    scan_kernel<<<scanGrid, 192, 0, stream>>>(x, part, cs);

    gather_kernel<<<dim3(N, B), 256, 0, stream>>>(x, spans, ptl, wtab, ptab,
                                                  cs, out, N);
}